// CenterLoss_7232724927176
// MI455X (gfx1250) — compile-verified
//
#include <hip/hip_runtime.h>
#include <stdint.h>

#define B_ROWS   16384
#define E_DIM    256
#define C_CLS    100000
#define ALPHA    0.5f
#define EPS_NRM  1e-12f

// ---------------------------------------------------------------------------
// Deterministic 256-thread block sum: wave32 shuffle butterfly, then an
// 8-entry LDS combine read uniformly by every thread (fixed order -> same
// bits on every call). scratch must hold >= 8 floats.
// ---------------------------------------------------------------------------
__device__ __forceinline__ float block_sum_256(float v, float* scratch) {
    #pragma unroll
    for (int m = 16; m > 0; m >>= 1) v += __shfl_xor(v, m, 32);
    const int lane = threadIdx.x & 31;
    const int wave = threadIdx.x >> 5;
    __syncthreads();                       // protect scratch across calls
    if (lane == 0) scratch[wave] = v;
    __syncthreads();
    float t = 0.0f;
    #pragma unroll
    for (int i = 0; i < 8; ++i) t += scratch[i];   // broadcast reads
    return t;
}

// ---------------------------------------------------------------------------
// K1: zero the per-class counts (tiny: 400 KB)
// ---------------------------------------------------------------------------
__global__ void zero_counts_kernel(float* __restrict__ counts) {
    int i = blockIdx.x * blockDim.x + threadIdx.x;
    if (i < C_CLS) counts[i] = 0.0f;
}

// ---------------------------------------------------------------------------
// K2: zero only the <=16384 labeled sums rows (instead of all 100000
// -> saves ~85 MB of zero+reread traffic). Duplicate labels benign.
// ---------------------------------------------------------------------------
__global__ void zero_label_rows_kernel(const int* __restrict__ labels,
                                       float* __restrict__ S) {
    int lab = labels[blockIdx.x];
    S[(size_t)lab * E_DIM + threadIdx.x] = 0.0f;
}

// ---------------------------------------------------------------------------
// K3: one 256-thread block per embedding row.
//  - row L2 norm (shuffle reduction)
//  - per-row MSE partial vs centers[label]*norm -> partials[b]
//  - scatter-add normalized embedding into sums rows + count
// ---------------------------------------------------------------------------
__global__ void row_kernel(const float* __restrict__ emb,
                           const float* __restrict__ centers,
                           const int*   __restrict__ labels,
                           float* __restrict__ S,
                           float* __restrict__ counts,
                           float* __restrict__ partials) {
    __shared__ float scratch[8];
    const int b   = blockIdx.x;
    const int tid = threadIdx.x;
    const int lab = labels[b];

    const float e   = emb[(size_t)b * E_DIM + tid];
    const float nrm = sqrtf(block_sum_256(e * e, scratch));   // ||emb_b||

    // loss partial: (e - centers[lab]*nrm)^2 summed over the row
    const float c = centers[(size_t)lab * E_DIM + tid];       // gather (L2-resident)
    const float d = e - c * nrm;
    const float l = block_sum_256(d * d, scratch);
    if (tid == 0) partials[b] = l;

    // scatter normalized embedding (reference divides without eps)
    atomicAdd(&S[(size_t)lab * E_DIM + tid], e / nrm);
    if (tid == 0) atomicAdd(&counts[lab], 1.0f);
}

// ---------------------------------------------------------------------------
// K4 (dominant: streams 102.4 MB in + 102.4 MB out): one block per class row.
// The centers row is staged into LDS with the CDNA5 async-tensor path:
// 64 lanes (two full waves, EXEC all-ones) each issue a
// GLOBAL_LOAD_ASYNC_TO_LDS_B128 (16B global -> 16B LDS, both aligned),
// tracked by ASYNCcnt; the counts load overlaps the transfer. The sums row
// (base d_out+1, only 4B-aligned) is read b32 and only when present
// (poison-safe); the result overwrites the sums/output region in place.
// ---------------------------------------------------------------------------
__global__ void finalize_kernel(const float* __restrict__ centers,
                                const float* __restrict__ counts,
                                float* __restrict__ S) {
    __shared__ float ctile[E_DIM];
    __shared__ float scratch[8];
    const int c   = blockIdx.x;
    const int tid = threadIdx.x;

    if (tid < 64) {
        // Generic shared-aperture pointer: low 32 bits == LDS byte offset
        // (flat->LDS truncation, ISA 10.2).
        const float* gptr = centers + (size_t)c * E_DIM + tid * 4;
        uint32_t lds_off  = (uint32_t)(uintptr_t)&ctile[tid * 4];
        asm volatile("global_load_async_to_lds_b128 %0, %1, off"
                     :: "v"(lds_off), "v"(gptr)
                     : "memory");
    }

    const float cnt = counts[c];          // overlaps with the async transfer

    asm volatile("s_wait_asynccnt 0x0" ::: "memory");
    __syncthreads();

    const float cv = ctile[tid];
    float out;
    if (cnt > 0.0f) {                     // uniform per block
        const float mean = S[(size_t)c * E_DIM + tid] / cnt;
        const float u    = cv - ALPHA * (cv - mean);          // = 0.5*(cv+mean)
        const float n2   = block_sum_256(u * u, scratch);
        out = u * (1.0f / fmaxf(sqrtf(n2), EPS_NRM));
    } else {
        out = cv;                         // absent class: pass centers through
    }
    S[(size_t)c * E_DIM + tid] = out;
}

// ---------------------------------------------------------------------------
// K5: deterministic loss reduction over the 16384 per-row partials.
// ---------------------------------------------------------------------------
__global__ void loss_kernel(const float* __restrict__ partials,
                            float* __restrict__ out_loss) {
    __shared__ float scratch[8];
    const int tid = threadIdx.x;
    float s = 0.0f;
    for (int i = tid; i < B_ROWS; i += 256) s += partials[i];
    const float total = block_sum_256(s, scratch);
    if (tid == 0)
        out_loss[0] = total * (1.0f / ((float)B_ROWS * (float)E_DIM));
}

// ---------------------------------------------------------------------------
extern "C" void kernel_launch(void* const* d_in, const int* in_sizes, int n_in,
                              void* d_out, int out_size, void* d_ws, size_t ws_size,
                              hipStream_t stream) {
    const float* emb     = (const float*)d_in[0];   // [16384,256] f32
    const float* centers = (const float*)d_in[1];   // [100000,256] f32
    const int*   labels  = (const int*)d_in[2];     // [16384] i32

    float* out  = (float*)d_out;
    float* loss = out;          // d_out[0]  : scalar center_loss
    float* S    = out + 1;      // d_out[1..]: sums scratch, then new_centers

    // workspace: counts[100000] + per-row loss partials[16384]  (~466 KB)
    float* counts   = (float*)d_ws;
    float* partials = counts + ((C_CLS + 127) & ~127);

    zero_counts_kernel    <<<(C_CLS + 255) / 256, 256, 0, stream>>>(counts);
    zero_label_rows_kernel<<<B_ROWS, E_DIM, 0, stream>>>(labels, S);
    row_kernel            <<<B_ROWS, E_DIM, 0, stream>>>(emb, centers, labels,
                                                         S, counts, partials);
    finalize_kernel       <<<C_CLS, E_DIM, 0, stream>>>(centers, counts, S);
    loss_kernel           <<<1, 256, 0, stream>>>(partials, loss);
}